// TransformerConv_29703993819985
// MI455X (gfx1250) — compile-verified
//
#include <hip/hip_runtime.h>

// Problem constants (match reference)
#define N_NODES 40000
#define N_EDGES 320000
#define D_IN    512
#define N_HEADS 4
#define D_OUT   64
#define D_MODEL (N_HEADS * D_OUT)   // 256

#define KC     64        // K-chunk staged in LDS per iteration
#define BS_LD  68        // padded LDS row stride (floats): bank-conflict-free b64 reads

typedef __attribute__((ext_vector_type(2))) float v2f;
typedef __attribute__((ext_vector_type(4))) float v4f;
typedef __attribute__((ext_vector_type(8))) float v8f;

// ---------------------------------------------------------------------------
// Kernel 0: init segment-max to -inf, segment-sum to 0
// ---------------------------------------------------------------------------
__global__ void tc_init_seg(float* __restrict__ segmax, float* __restrict__ segsum) {
  int i = blockIdx.x * blockDim.x + threadIdx.x;
  if (i < N_NODES * N_HEADS) {
    segmax[i] = -__builtin_huge_valf();
    segsum[i] = 0.0f;
  }
}

// ---------------------------------------------------------------------------
// Kernel 1: ft = feat @ W with V_WMMA_F32_16X16X4_F32 (wave32 tensor path).
//
// Workgroup = 128 threads = 4 waves. All 4 waves share one 64-wide N strip;
// each wave owns one 16-row M tile (4 tiles / block). The W strip for a
// 64-deep K chunk is staged in LDS transposed (Bs[n][k], stride BS_LD) so a
// B fragment is ONE ds_load_b64 per 16x16 subtile.
//   A 16x4 : lane L -> row M = L&15, K = 2*(L>>4)+{0,1}  (v2f, global b64)
//   B 4x16 : lane L -> col N = L&15, K = 2*(L>>4)+{0,1}  (v2f, LDS b64)
//   C 16x16: VGPR r, lane L -> (M = r + 8*(L>>4), N = L&15)   (v8f)
// Staging is two-phase (global->regs, regs->LDS) so loads clause-batch with
// one wait; the compute loop register-double-buffers A/B fragments so DS/VMEM
// latency hides behind the WMMA pipe. Also seeds out = ft (residual term).
// ---------------------------------------------------------------------------
__global__ __launch_bounds__(128) void tc_gemm_wmma_f32(
    const float* __restrict__ feat,   // [N_NODES, D_IN] row-major
    const float* __restrict__ W,      // [D_IN, D_MODEL] row-major
    float* __restrict__ ft,           // [N_NODES, D_MODEL]
    float* __restrict__ out) {        // [N_NODES, D_MODEL]
  __shared__ float Bs[64 * BS_LD];    // 64 N-cols x (KC + pad)  ~17.4 KB

  const int tid   = threadIdx.x;
  const int wavei = tid >> 5;                 // 0..3 : M tile within block
  const int lane  = tid & 31;
  const int mgrp  = blockIdx.x >> 2;          // 0..624
  const int n0    = (blockIdx.x & 3) * 64;    // N strip: 0,64,128,192
  const int m0    = (mgrp * 4 + wavei) * 16;  // 2500 M tiles, exact
  const int mrow  = lane & 15;
  const int ncol  = lane & 15;
  const int kgrp  = (lane >> 4) * 2;          // 0 or 2

  const float* __restrict__ Arow = feat + (size_t)(m0 + mrow) * D_IN + kgrp;

  v8f acc0 = {}, acc1 = {}, acc2 = {}, acc3 = {};

  const int sc = tid & 63;          // staging: column within strip
  const int sr = tid >> 6;          // staging: row parity (0/1)

  for (int kc = 0; kc < D_IN; kc += KC) {
    // ---- stage W[kc..kc+63][n0..n0+63] into LDS, transposed to Bs[n][k] ----
    // phase 1: batch all global loads into registers (one wait, clause-batched)
    float wreg[KC / 2];
    #pragma unroll
    for (int i = 0; i < KC / 2; ++i) {
      int r = i * 2 + sr;                                   // K row in chunk
      wreg[i] = W[(size_t)(kc + r) * D_MODEL + n0 + sc];
    }
    // phase 2: batch all DS stores
    #pragma unroll
    for (int i = 0; i < KC / 2; ++i) {
      int r = i * 2 + sr;
      Bs[sc * BS_LD + r] = wreg[i];
    }
    __syncthreads();

    // ---- 16 WMMA k-steps out of LDS, register-double-buffered ----
    const float* bbase = &Bs[ncol * BS_LD + kgrp];
    v2f a0 = *(const v2f*)(Arow + kc);
    v2f b0 = *(const v2f*)(bbase);
    v2f b1 = *(const v2f*)(bbase + 16 * BS_LD);
    v2f b2 = *(const v2f*)(bbase + 32 * BS_LD);
    v2f b3 = *(const v2f*)(bbase + 48 * BS_LD);

    #pragma unroll
    for (int kk = 0; kk < KC; kk += 4) {
      v2f an = {}, bn0 = {}, bn1 = {}, bn2 = {}, bn3 = {};
      if (kk + 4 < KC) {                       // prefetch next k-step
        an  = *(const v2f*)(Arow + kc + kk + 4);
        const float* bp = bbase + kk + 4;
        bn0 = *(const v2f*)(bp);
        bn1 = *(const v2f*)(bp + 16 * BS_LD);
        bn2 = *(const v2f*)(bp + 32 * BS_LD);
        bn3 = *(const v2f*)(bp + 48 * BS_LD);
      }
      acc0 = __builtin_amdgcn_wmma_f32_16x16x4_f32(false, a0, false, b0, (short)0, acc0, false, false);
      acc1 = __builtin_amdgcn_wmma_f32_16x16x4_f32(false, a0, false, b1, (short)0, acc1, false, false);
      acc2 = __builtin_amdgcn_wmma_f32_16x16x4_f32(false, a0, false, b2, (short)0, acc2, false, false);
      acc3 = __builtin_amdgcn_wmma_f32_16x16x4_f32(false, a0, false, b3, (short)0, acc3, false, false);
      a0 = an; b0 = bn0; b1 = bn1; b2 = bn2; b3 = bn3;
    }
    __syncthreads();
  }

  const int rbase = 8 * (lane >> 4);
  #pragma unroll
  for (int r = 0; r < 8; ++r) {
    size_t row = (size_t)(m0 + r + rbase) * D_MODEL;
    float v0 = acc0[r], v1 = acc1[r], v2 = acc2[r], v3 = acc3[r];
    ft[row + n0 +  0 + ncol] = v0;  out[row + n0 +  0 + ncol] = v0;
    ft[row + n0 + 16 + ncol] = v1;  out[row + n0 + 16 + ncol] = v1;
    ft[row + n0 + 32 + ncol] = v2;  out[row + n0 + 32 + ncol] = v2;
    ft[row + n0 + 48 + ncol] = v3;  out[row + n0 + 48 + ncol] = v3;
  }
}

// ---------------------------------------------------------------------------
// Ordered-int float atomic max (works for all finite floats, init = -inf)
// ---------------------------------------------------------------------------
__device__ inline void atomicMaxF(float* addr, float val) {
  if (val >= 0.0f)
    atomicMax((int*)addr, __float_as_int(val));
  else
    atomicMin((unsigned int*)addr, __float_as_uint(val));
}

// ---------------------------------------------------------------------------
// Kernel 2: per-(edge,head) scaled dot-product logits + segment max over dst.
// ft rows are L2-resident (41 MB << 192 MB L2) so gathers hit L2.
// ---------------------------------------------------------------------------
__global__ void tc_edge_logits(
    const float* __restrict__ ft,
    const int* __restrict__ src, const int* __restrict__ dst,
    float* __restrict__ logits, float* __restrict__ segmax) {
  int idx = blockIdx.x * blockDim.x + threadIdx.x;
  if (idx >= N_EDGES * N_HEADS) return;
  int e = idx >> 2, h = idx & 3;
  int s = src[e], d = dst[e];
  const float* __restrict__ kp = ft + (size_t)s * D_MODEL + h * D_OUT;
  const float* __restrict__ qp = ft + (size_t)d * D_MODEL + h * D_OUT;
  float dot = 0.0f;
  #pragma unroll
  for (int i = 0; i < D_OUT; i += 4) {
    v4f kv = *(const v4f*)(kp + i);
    v4f qv = *(const v4f*)(qp + i);
    dot += kv.x * qv.x + kv.y * qv.y + kv.z * qv.z + kv.w * qv.w;
  }
  float logit = dot * 0.125f;                 // 1/sqrt(64)
  logits[idx] = logit;
  atomicMaxF(&segmax[d * N_HEADS + h], logit);
}

// ---------------------------------------------------------------------------
// Kernel 3: ex = exp(logit - segmax[dst]); segment sum over dst.
// ---------------------------------------------------------------------------
__global__ void tc_edge_expsum(
    const int* __restrict__ dst,
    const float* __restrict__ logits, const float* __restrict__ segmax,
    float* __restrict__ ex, float* __restrict__ segsum) {
  int idx = blockIdx.x * blockDim.x + threadIdx.x;
  if (idx >= N_EDGES * N_HEADS) return;
  int e = idx >> 2, h = idx & 3;
  int d = dst[e];
  float v = __expf(logits[idx] - segmax[d * N_HEADS + h]);
  ex[idx] = v;
  atomicAdd(&segsum[d * N_HEADS + h], v);
}

// ---------------------------------------------------------------------------
// Kernel 4: out[dst] += (ex/segsum[dst]) * ft[src], one 256-thread block per
// edge: coalesced row gather + coalesced global_atomic_add_f32 into out row.
// ---------------------------------------------------------------------------
__global__ __launch_bounds__(256) void tc_edge_aggregate(
    const float* __restrict__ ft,
    const int* __restrict__ src, const int* __restrict__ dst,
    const float* __restrict__ ex, const float* __restrict__ segsum,
    float* __restrict__ out) {
  int e = blockIdx.x;
  int j = threadIdx.x;            // 0..255 : h = j>>6, d = j&63
  int h = j >> 6;
  int s = src[e], d = dst[e];
  float coeff = ex[e * N_HEADS + h] / segsum[d * N_HEADS + h];
  float key   = ft[(size_t)s * D_MODEL + j];
  atomicAdd(&out[(size_t)d * D_MODEL + j], coeff * key);
}

// ---------------------------------------------------------------------------
// Host-side launch
// ---------------------------------------------------------------------------
extern "C" void kernel_launch(void* const* d_in, const int* in_sizes, int n_in,
                              void* d_out, int out_size, void* d_ws, size_t ws_size,
                              hipStream_t stream) {
  const float* feat = (const float*)d_in[0];   // [40000, 512]
  const float* W    = (const float*)d_in[1];   // [512, 256]
  const int*   src  = (const int*)d_in[2];     // [320000]
  const int*   dst  = (const int*)d_in[3];     // [320000]
  float*       out  = (float*)d_out;           // [40000, 256]

  // workspace layout (floats)
  float* ws     = (float*)d_ws;
  float* ft     = ws;                                   // 40000*256 = 10,240,000
  float* logits = ft + (size_t)N_NODES * D_MODEL;       // 1,280,000
  float* ex     = logits + (size_t)N_EDGES * N_HEADS;   // 1,280,000
  float* segmax = ex + (size_t)N_EDGES * N_HEADS;       //   160,000
  float* segsum = segmax + (size_t)N_NODES * N_HEADS;   //   160,000

  // 0) init segment buffers
  tc_init_seg<<<(N_NODES * N_HEADS + 255) / 256, 256, 0, stream>>>(segmax, segsum);

  // 1) ft = feat @ W (fp32 WMMA, LDS-staged B), out = ft
  // grid = 625 M-groups x 4 N-strips; 128 threads = 4 waves per block
  tc_gemm_wmma_f32<<<2500, 128, 0, stream>>>(feat, W, ft, out);

  // 2) logits + segment max
  tc_edge_logits<<<(N_EDGES * N_HEADS + 255) / 256, 256, 0, stream>>>(
      ft, src, dst, logits, segmax);

  // 3) exp + segment sum
  tc_edge_expsum<<<(N_EDGES * N_HEADS + 255) / 256, 256, 0, stream>>>(
      dst, logits, segmax, ex, segsum);

  // 4) weighted aggregation into out
  tc_edge_aggregate<<<N_EDGES, 256, 0, stream>>>(ft, src, dst, ex, segsum, out);
}